// NCC_118
// MI455X (gfx1250) — compile-verified
//
#include <hip/hip_runtime.h>

// ---------------------------------------------------------------------------
// Local NCC loss, fused single-pass tile kernel for MI455X (gfx1250, wave32).
//
//   * tile = 8(z) x 16(y) x 16(x) outputs, halo 4 -> 16x24x24 inputs in LDS
//   * z-pass / y-pass: sliding-window 9-tap sums in LDS
//   * x-pass: exact fp32 WMMA 16x16x4 against banded ones matrices
//   * loss fused in registers, deterministic 2-stage reduction via d_ws
// ---------------------------------------------------------------------------

typedef float v2f __attribute__((ext_vector_type(2)));
typedef float v8f __attribute__((ext_vector_type(8)));

#if defined(__has_builtin)
#if __has_builtin(__builtin_amdgcn_wmma_f32_16x16x4_f32)
#define NCC_HAVE_WMMA_F32 1
#endif
#endif
#ifndef NCC_HAVE_WMMA_F32
#define NCC_HAVE_WMMA_F32 0
#endif

namespace {
constexpr int D = 160, H = 192, W = 224;   // volume (z, y, x)
constexpr int TZ = 8, TY = 16, TX = 16;    // output tile
constexpr int RZ = TZ + 8, RY = TY + 8, RX = TX + 8;  // haloed input tile
constexpr int XP = 25;                     // padded x stride (bank-conflict free)
constexpr float WIN_INV = 1.0f / 729.0f;
}

__device__ __forceinline__ float fieldVal(int f, float a, float b) {
  switch (f) {
    case 0:  return a;
    case 1:  return b;
    case 2:  return a * a;
    case 3:  return b * b;
    default: return a * b;
  }
}

__global__ __launch_bounds__(256)
void ncc_tile_kernel(const float* __restrict__ gI, const float* __restrict__ gJ,
                     float* __restrict__ partial) {
  __shared__ float rawI[RZ][RY][XP];   // 16*24*25*4 = 38400 B
  __shared__ float rawJ[RZ][RY][XP];   // 38400 B
  __shared__ float bufA[TZ][RY][XP];   //  8*24*25*4 = 19200 B  (z-filtered)
  __shared__ float bufB[TZ][TY][XP];   //  8*16*25*4 = 12800 B  (z+y filtered)
  __shared__ float wsum[8];

  const int tid = threadIdx.x;
  const int z0 = blockIdx.z * TZ;
  const int y0 = blockIdx.y * TY;
  const int x0 = blockIdx.x * TX;

  // ---- Stage 0: cooperative haloed load (zero padding outside the volume) --
  for (int idx = tid; idx < RZ * RY * RX; idx += 256) {
    int lx = idx % RX;
    int t  = idx / RX;
    int ly = t % RY;
    int lz = t / RY;
    int gz = z0 - 4 + lz, gy = y0 - 4 + ly, gx = x0 - 4 + lx;
    bool ok = (unsigned)gz < (unsigned)D && (unsigned)gy < (unsigned)H &&
              (unsigned)gx < (unsigned)W;
    size_t g = ((size_t)gz * H + gy) * (size_t)W + gx;
    rawI[lz][ly][lx] = ok ? gI[g] : 0.0f;
    rawJ[lz][ly][lx] = ok ? gJ[g] : 0.0f;
  }
  __syncthreads();

  const int lane = tid & 31;
  const int wave = tid >> 5;          // 8 waves -> one z-slice (WMMA tile) each
  const int m    = lane & 15;         // row (y) for A, column (x=n) for B/D
  const int kb   = (lane >> 4) << 1;  // K base: lanes 0-15 -> {0,1}; 16-31 -> {2,3}

  v8f S[5];                           // per-thread window sums for the 5 fields

#pragma unroll
  for (int f = 0; f < 5; ++f) {
    // ---- z-pass: 9-tap sliding sum along z (576 (y,x) columns) -------------
    for (int col = tid; col < RY * RX; col += 256) {
      int x = col % RX, y = col / RX;
      float s = 0.0f;
#pragma unroll
      for (int lz = 0; lz < 9; ++lz)
        s += fieldVal(f, rawI[lz][y][x], rawJ[lz][y][x]);
      bufA[0][y][x] = s;
#pragma unroll
      for (int zo = 1; zo < TZ; ++zo) {
        s += fieldVal(f, rawI[zo + 8][y][x], rawJ[zo + 8][y][x]) -
             fieldVal(f, rawI[zo - 1][y][x], rawJ[zo - 1][y][x]);
        bufA[zo][y][x] = s;
      }
    }
    __syncthreads();

    // ---- y-pass: 9-tap sliding sum along y (192 (z,x) columns) -------------
    if (tid < TZ * RX) {
      int x = tid % RX, z = tid / RX;
      float s = 0.0f;
#pragma unroll
      for (int ly = 0; ly < 9; ++ly) s += bufA[z][ly][x];
      bufB[z][0][x] = s;
#pragma unroll
      for (int yo = 1; yo < TY; ++yo) {
        s += bufA[z][yo + 8][x] - bufA[z][yo - 1][x];
        bufB[z][yo][x] = s;
      }
    }
    __syncthreads();

    // ---- x-pass: banded ones matmul via fp32 WMMA (exact) ------------------
    // out(16y x 16x) = sum_{c=0..5} A_c(16x4) * W_c(4x16),
    //   A_c[m][k] = bufB[z][m][4c+k],  W_c[k][n] = 1 iff 0 <= 4c+k-n <= 8
    v8f acc = {0.f, 0.f, 0.f, 0.f, 0.f, 0.f, 0.f, 0.f};
#if NCC_HAVE_WMMA_F32
#pragma unroll
    for (int c = 0; c < 6; ++c) {
      v2f a, b;
      a.x = bufB[wave][m][4 * c + kb];
      a.y = bufB[wave][m][4 * c + kb + 1];
      int d0 = 4 * c + kb - m;                    // n == lane&15 == m
      b.x = (d0 >= 0 && d0 <= 8) ? 1.0f : 0.0f;
      b.y = (d0 + 1 >= 0 && d0 + 1 <= 8) ? 1.0f : 0.0f;
      acc = __builtin_amdgcn_wmma_f32_16x16x4_f32(
          /*neg_a=*/false, a, /*neg_b=*/false, b,
          /*c_mod=*/(short)0, acc, /*reuse_a=*/false, /*reuse_b=*/false);
    }
#else
    // Scalar fallback with identical register->voxel mapping (D layout).
#pragma unroll
    for (int r = 0; r < 8; ++r) {
      int yy = r + ((lane >> 4) << 3);
      float s = 0.0f;
#pragma unroll
      for (int t2 = 0; t2 < 9; ++t2) s += bufB[wave][yy][m + t2];
      acc[r] = s;
    }
#endif
    S[f] = acc;
    __syncthreads();  // bufA/bufB reused by the next field
  }

  // ---- Fused loss: D-layout register r -> voxel (z=wave, y=r+8*(lane>=16), x=m)
  float local = 0.0f;
#pragma unroll
  for (int r = 0; r < 8; ++r) {
    float sI  = S[0][r], sJ  = S[1][r];
    float sI2 = S[2][r], sJ2 = S[3][r], sIJ = S[4][r];
    float cross = sIJ - sI * sJ * WIN_INV;   // == IJ_sum - u_J*I_sum - u_I*J_sum + u_I*u_J*729
    float Ivar  = sI2 - sI * sI * WIN_INV;
    float Jvar  = sJ2 - sJ * sJ * WIN_INV;
    local += cross * cross / (Ivar * Jvar + 1e-5f);
  }

  // ---- Deterministic in-block reduction ------------------------------------
  for (int off = 16; off > 0; off >>= 1) local += __shfl_down(local, off, 32);
  if (lane == 0) wsum[wave] = local;
  __syncthreads();
  if (tid == 0) {
    float s = 0.0f;
#pragma unroll
    for (int w = 0; w < 8; ++w) s += wsum[w];
    partial[blockIdx.x + gridDim.x * (blockIdx.y + gridDim.y * blockIdx.z)] = s;
  }
}

__global__ __launch_bounds__(256)
void ncc_reduce_kernel(const float* __restrict__ partial, int n,
                       float* __restrict__ out, float inv_count) {
  __shared__ float sm[256];
  float s = 0.0f;
  for (int i = threadIdx.x; i < n; i += 256) s += partial[i];
  sm[threadIdx.x] = s;
  __syncthreads();
  for (int st = 128; st > 0; st >>= 1) {
    if (threadIdx.x < st) sm[threadIdx.x] += sm[threadIdx.x + st];
    __syncthreads();
  }
  if (threadIdx.x == 0) out[0] = -sm[0] * inv_count;  // loss = -mean(cc)
}

extern "C" void kernel_launch(void* const* d_in, const int* in_sizes, int n_in,
                              void* d_out, int out_size, void* d_ws, size_t ws_size,
                              hipStream_t stream) {
  (void)in_sizes; (void)n_in; (void)out_size; (void)ws_size;
  const float* gI = (const float*)d_in[0];  // y_true
  const float* gJ = (const float*)d_in[1];  // y_pred
  float* partial = (float*)d_ws;            // 3360 floats of scratch

  dim3 grid(W / TX, H / TY, D / TZ);        // 14 x 12 x 20 = 3360 blocks
  ncc_tile_kernel<<<grid, 256, 0, stream>>>(gI, gJ, partial);

  const int nblocks = grid.x * grid.y * grid.z;
  const float inv_count = 1.0f / (float)((size_t)D * H * W);
  ncc_reduce_kernel<<<1, 256, 0, stream>>>(partial, nblocks, (float*)d_out, inv_count);
}